// SimpleGNN_37701222924601
// MI455X (gfx1250) — compile-verified
//
#include <hip/hip_runtime.h>
#include <math.h>

typedef __attribute__((ext_vector_type(2))) float v2f;
typedef __attribute__((ext_vector_type(8))) float v8f;

#define N_NODES 100000
#define N_EDGES 1600000
#define N_EGO   8192
#define HIDDEN  128
#define KPAD    132   // 129 padded up to a multiple of 4 for the K=4 WMMA
#define H1DIM   64
#define OUTDIM  16

// workspace layout, offsets in floats (all 1KB-aligned)
#define O_DINV   0u                       // N_NODES floats (deg, then rsqrt in place)
#define O_NEEDED 100352u                  // N_NODES ints
#define O_AGG    200704u                  // N_NODES*128 floats
#define O_H      13000704u                // N_EGO*KPAD floats
#define O_W1P    14082048u                // KPAD*H1DIM floats
#define O_H1     14090496u                // N_EGO*H1DIM floats  (ends ~58.5 MB)

// ---- init: deg=1 (self loop), needed=0, agg=0 ----------------------------
__global__ void k_init(float* __restrict__ deg, int* __restrict__ needed,
                       float* __restrict__ agg) {
  int tid = blockIdx.x * blockDim.x + threadIdx.x;   // N_NODES*32 threads
  int n = tid >> 5, j = tid & 31;
  if (n >= N_NODES) return;
  *(float4*)(agg + (size_t)n * HIDDEN + 4 * j) = make_float4(0.f, 0.f, 0.f, 0.f);
  if (j == 0) { deg[n] = 1.0f; needed[n] = 0; }
}

__global__ void k_mark(const int* __restrict__ ego_idx, int* __restrict__ needed) {
  int i = blockIdx.x * blockDim.x + threadIdx.x;
  if (i < N_EGO) needed[ego_idx[i]] = 1;
}

__global__ void k_deg(const int* __restrict__ edge, float* __restrict__ deg) {
  int e = blockIdx.x * blockDim.x + threadIdx.x;
  if (e < N_EDGES) atomicAdd(&deg[edge[N_EDGES + e]], 1.0f);
}

__global__ void k_dinv(float* __restrict__ deg) {
  int n = blockIdx.x * blockDim.x + threadIdx.x;
  if (n < N_NODES) deg[n] = rsqrtf(deg[n]);   // deg >= 1 always (self loop)
}

// ---- edge scatter: one wave per edge; skip edges whose dst is not an ego node
__global__ void k_scatter(const int* __restrict__ edge, const float* __restrict__ data1,
                          const float* __restrict__ conv_W, const float* __restrict__ dinv,
                          const int* __restrict__ needed, float* __restrict__ agg) {
  int wave = blockIdx.x * (blockDim.x >> 5) + (threadIdx.x >> 5);
  int lane = threadIdx.x & 31;
  if (wave >= N_EDGES) return;
  int dst = edge[N_EDGES + wave];
  if (!needed[dst]) return;                       // ~92% of waves exit here
  int src = edge[wave];
  float norm = dinv[src] * dinv[dst];
  float4 d = *(const float4*)(data1 + 4 * (size_t)src);
  const float* wb = conv_W + 4 * lane;            // lane covers features 4l..4l+3
  float4 w0 = *(const float4*)(wb);
  float4 w1 = *(const float4*)(wb + 128);
  float4 w2 = *(const float4*)(wb + 256);
  float4 w3 = *(const float4*)(wb + 384);
  float* a = agg + (size_t)dst * HIDDEN + 4 * lane;
  atomicAdd(a + 0, (d.x * w0.x + d.y * w1.x + d.z * w2.x + d.w * w3.x) * norm);
  atomicAdd(a + 1, (d.x * w0.y + d.y * w1.y + d.z * w2.y + d.w * w3.y) * norm);
  atomicAdd(a + 2, (d.x * w0.z + d.y * w1.z + d.z * w2.z + d.w * w3.z) * norm);
  atomicAdd(a + 3, (d.x * w0.w + d.y * w1.w + d.z * w2.w + d.w * w3.w) * norm);
}

// ---- gather ego rows, add self-loop + bias, concat time, pad K to 132 ----
__global__ void k_gather(const int* __restrict__ ego_idx, const float* __restrict__ ego_time,
                         const float* __restrict__ data1, const float* __restrict__ conv_W,
                         const float* __restrict__ conv_b, const float* __restrict__ dinv,
                         const float* __restrict__ agg, float* __restrict__ H) {
  int idx = blockIdx.x * blockDim.x + threadIdx.x;   // N_EGO*KPAD threads
  int i = idx / KPAD, c = idx % KPAD;
  if (i >= N_EGO) return;
  float val;
  if (c < HIDDEN) {
    int v = ego_idx[i];
    float di = dinv[v];
    float4 d = *(const float4*)(data1 + 4 * (size_t)v);
    float self = d.x * conv_W[c] + d.y * conv_W[HIDDEN + c] +
                 d.z * conv_W[2 * HIDDEN + c] + d.w * conv_W[3 * HIDDEN + c];
    val = agg[(size_t)v * HIDDEN + c] + self * di * di + conv_b[c];
  } else if (c == HIDDEN) {
    val = ego_time[i];
  } else {
    val = 0.0f;                                    // K padding
  }
  H[(size_t)i * KPAD + c] = val;
}

// ---- zero-pad lin1_W from 129x64 to 132x64 --------------------------------
__global__ void k_w1pad(const float* __restrict__ lin1_W, float* __restrict__ W1p) {
  int idx = blockIdx.x * blockDim.x + threadIdx.x;   // KPAD*H1DIM threads
  int r = idx / H1DIM;
  if (r >= KPAD) return;
  W1p[idx] = (r < HIDDEN + 1) ? lin1_W[idx] : 0.0f;
}

// ---- lin1: [8192,132] @ [132,64] via V_WMMA_F32_16X16X4_F32, ReLU+bias ----
__global__ void k_lin1(const float* __restrict__ H, const float* __restrict__ W1p,
                       const float* __restrict__ b1, float* __restrict__ h1) {
  int wave = blockIdx.x * (blockDim.x >> 5) + (threadIdx.x >> 5); // 2048 waves
  int lane = threadIdx.x & 31;
  int tm = wave >> 2, tn = wave & 3;
  int lr = lane & 15;
  int ksel = (lane >> 4) << 1;                   // lanes 0-15 -> K 0,1 ; 16-31 -> K 2,3
  int m = tm * 16 + lr;
  int n = tn * 16 + lr;
  const float* hrow = H + (size_t)m * KPAD;
  v8f c = {};
  #pragma unroll
  for (int k0 = 0; k0 < KPAD; k0 += 4) {
    v2f a = *(const v2f*)(hrow + k0 + ksel);
    v2f b;
    b.x = W1p[(k0 + ksel) * H1DIM + n];
    b.y = W1p[(k0 + ksel + 1) * H1DIM + n];
    c = __builtin_amdgcn_wmma_f32_16x16x4_f32(false, a, false, b,
                                              (short)0, c, false, false);
  }
  float bias = b1[n];
  int mbase = tm * 16 + ((lane >> 4) << 3);      // C layout: M = r + 8*(lane>=16)
  #pragma unroll
  for (int r = 0; r < 8; r++) {
    float v = c[r] + bias;
    h1[(size_t)(mbase + r) * H1DIM + n] = v > 0.f ? v : 0.f;
  }
}

// ---- lin2: [8192,64] @ [64,16] via WMMA, bias + sigmoid -> d_out ----------
__global__ void k_lin2(const float* __restrict__ h1, const float* __restrict__ W2,
                       const float* __restrict__ b2, float* __restrict__ out) {
  int wave = blockIdx.x * (blockDim.x >> 5) + (threadIdx.x >> 5); // 512 waves
  int lane = threadIdx.x & 31;
  int lr = lane & 15;
  int ksel = (lane >> 4) << 1;
  int m = wave * 16 + lr;
  int n = lr;                                    // single N tile (OUT=16)
  const float* hrow = h1 + (size_t)m * H1DIM;
  v8f c = {};
  #pragma unroll
  for (int k0 = 0; k0 < H1DIM; k0 += 4) {
    v2f a = *(const v2f*)(hrow + k0 + ksel);
    v2f b;
    b.x = W2[(k0 + ksel) * OUTDIM + n];
    b.y = W2[(k0 + ksel + 1) * OUTDIM + n];
    c = __builtin_amdgcn_wmma_f32_16x16x4_f32(false, a, false, b,
                                              (short)0, c, false, false);
  }
  float bias = b2[n];
  int mbase = wave * 16 + ((lane >> 4) << 3);
  #pragma unroll
  for (int r = 0; r < 8; r++) {
    float v = c[r] + bias;
    out[(size_t)(mbase + r) * OUTDIM + n] = 1.0f / (1.0f + __expf(-v));
  }
}

extern "C" void kernel_launch(void* const* d_in, const int* in_sizes, int n_in,
                              void* d_out, int out_size, void* d_ws, size_t ws_size,
                              hipStream_t stream) {
  (void)in_sizes; (void)n_in; (void)out_size; (void)ws_size;
  const float* data1    = (const float*)d_in[0];
  const int*   edge     = (const int*)d_in[1];
  const float* ego_time = (const float*)d_in[2];
  const int*   ego_idx  = (const int*)d_in[3];
  const float* conv_W   = (const float*)d_in[4];
  const float* conv_b   = (const float*)d_in[5];
  const float* lin1_W   = (const float*)d_in[6];
  const float* lin1_b   = (const float*)d_in[7];
  const float* lin2_W   = (const float*)d_in[8];
  const float* lin2_b   = (const float*)d_in[9];
  float* out = (float*)d_out;

  float* ws     = (float*)d_ws;
  float* dinv   = ws + O_DINV;
  int*   needed = (int*)(ws + O_NEEDED);
  float* agg    = ws + O_AGG;
  float* H      = ws + O_H;
  float* W1p    = ws + O_W1P;
  float* h1     = ws + O_H1;

  // init deg/needed/agg
  k_init<<<(N_NODES * 32 + 255) / 256, 256, 0, stream>>>(dinv, needed, agg);
  k_mark<<<(N_EGO + 255) / 256, 256, 0, stream>>>(ego_idx, needed);
  // degree over all edges, then rsqrt in place
  k_deg<<<(N_EDGES + 255) / 256, 256, 0, stream>>>(edge, dinv);
  k_dinv<<<(N_NODES + 255) / 256, 256, 0, stream>>>(dinv);
  // edge scatter (wave per edge, early-exit on non-ego dst)
  k_scatter<<<N_EDGES / 8, 256, 0, stream>>>(edge, data1, conv_W, dinv, needed, agg);
  // gather + self-loop + time concat -> H [8192 x 132]
  k_gather<<<(N_EGO * KPAD) / 256, 256, 0, stream>>>(ego_idx, ego_time, data1, conv_W,
                                                     conv_b, dinv, agg, H);
  // pad W1, then the two WMMA GEMMs
  k_w1pad<<<(KPAD * H1DIM) / 256, 256, 0, stream>>>(lin1_W, W1p);
  k_lin1<<<256, 256, 0, stream>>>(H, W1p, lin1_b, h1);   // 2048 waves, 8/block
  k_lin2<<<64, 256, 0, stream>>>(h1, lin2_W, lin2_b, out); // 512 waves
}